// PathQualityNetwork_69741678952849
// MI455X (gfx1250) — compile-verified
//
#include <hip/hip_runtime.h>
#include <hip/hip_bf16.h>

// ---------------------------------------------------------------------------
// PathQualityNetwork, fully fused for MI455X (gfx1250, wave32).
//
// Key identity: path-layer output columns are scaled columns of W, so the
// scorer factorizes through C = s_w1 @ W (computed with v_wmma_f32_16x16x4_f32
// per workgroup). Per-path state is (feature_index, scalar_value); candidates
// are scored with a 32-wide scalar MLP; exact top-100 via LDS radix select.
// One workgroup (256 threads = 8 wave32) per batch element; grid = 16.
// ---------------------------------------------------------------------------

typedef float v2f __attribute__((ext_vector_type(2)));
typedef float v8f __attribute__((ext_vector_type(8)));

#define BLOCK   256
#define NFEAT   256
#define NPATH   100
#define NCAND   (NPATH * NFEAT)   // 25600 candidates at layers 1,2
#define FOUT_LAST 128

// -------------------- dynamic LDS layout (bytes) ---------------------------
#define OFF_CT      0                        // float Ct[256][32]  (C transposed)
#define OFF_KEYS    (OFF_CT   + 32768)       // uint  keys[25600]
#define OFF_B1      (OFF_KEYS + 102400)      // float b1s[32]
#define OFF_W2S     (OFF_B1   + 128)         // float w2s[32]
#define OFF_PVAL    (OFF_W2S  + 128)         // float pval[100]
#define OFF_NVAL    (OFF_PVAL + 400)         // float nval[100]
#define OFF_NSC     (OFF_NVAL + 400)         // float nsc[100]
#define OFF_G       (OFF_NSC  + 400)         // float g[256]
#define OFF_PIDX    (OFF_G    + 1024)        // int   pidx[100]
#define OFF_NIDX    (OFF_PIDX + 400)         // int   nidx[100]
#define OFF_HIST    (OFF_NIDX + 400)         // uint  hist[256]
#define OFF_CTRL    (OFF_HIST + 1024)        // int   ctrl[4]: prefix,kk,tie,slot
#define SMEM_BYTES  (OFF_CTRL + 16)          // = 139488 B  (< 320 KB/WGP)

// -------------------- order-preserving float <-> uint key ------------------
__device__ __forceinline__ unsigned key_of(float f) {
  unsigned u = __float_as_uint(f);
  return (u & 0x80000000u) ? ~u : (u | 0x80000000u);
}
__device__ __forceinline__ float val_of(unsigned k) {
  unsigned v = (k & 0x80000000u) ? (k & 0x7fffffffu) : ~k;
  return __uint_as_float(v);
}

// -------------------- C^T = (s_w1 @ W)^T via f32 WMMA ----------------------
// M=32, N=256, K=Fout.  Output tiles: 2 x 16 = 32, four per wave (8 waves).
// A (16x4) lane layout: lanes 0-15 hold K=0,1; lanes 16-31 hold K=2,3.
// B (4x16): VGPR0: lanes0-15 K=0 / lanes16-31 K=2; VGPR1: K=1 / K=3.
// D (16x16 f32): VGPR r: lanes0-15 M=r, lanes16-31 M=r+8; N=lane&15.
__device__ void compute_Ct(const float* __restrict__ sw1,
                           const float* __restrict__ W,
                           int Fout, float* __restrict__ Ct) {
  const int tid  = threadIdx.x;
  const int lane = tid & 31;
  const int wave = tid >> 5;
  const int ml   = lane & 15;
  const int hi   = lane >> 4;
  for (int j = 0; j < 4; ++j) {
    const int t  = wave * 4 + j;          // tile 0..31
    const int m0 = (t >> 4) * 16;         // 0 or 16
    const int n0 = (t & 15) * 16;         // 0..240
    v8f acc = {0.f, 0.f, 0.f, 0.f, 0.f, 0.f, 0.f, 0.f};
    for (int k0 = 0; k0 < Fout; k0 += 4) {
      const int ka = k0 + hi * 2;
      v2f a, b;
      a.x = sw1[(m0 + ml) * Fout + ka];
      a.y = sw1[(m0 + ml) * Fout + ka + 1];
      b.x = W[ka * NFEAT + n0 + ml];
      b.y = W[(ka + 1) * NFEAT + n0 + ml];
      acc = __builtin_amdgcn_wmma_f32_16x16x4_f32(
          /*neg_a=*/false, a, /*neg_b=*/false, b,
          /*c_mod=*/(short)0, acc, /*reuse_a=*/false, /*reuse_b=*/false);
    }
#pragma unroll
    for (int r = 0; r < 8; ++r) {
      const int M = m0 + r + hi * 8;
      const int N = n0 + ml;
      Ct[N * 32 + M] = acc[r];
    }
  }
}

// -------------------- 32-wide scalar MLP score -----------------------------
__device__ __forceinline__ float mlp_score(const float* __restrict__ c, float s,
                                           const float* __restrict__ b1,
                                           const float* __restrict__ w2,
                                           float b2) {
  float acc = b2;
#pragma unroll
  for (int k = 0; k < 32; ++k) {
    float h = fmaf(c[k], s, b1[k]);
    acc = fmaf(fmaxf(h, 0.0f), w2[k], acc);
  }
  return acc;
}

// -------------------- exact top-K threshold via 4-pass radix select --------
// Result: ctrl[0] = threshold key T (keys > T all selected),
//         ctrl[1] = number of keys == T still to accept.
__device__ void radix_select(unsigned* __restrict__ keys, int N, int K,
                             unsigned* __restrict__ hist, int* __restrict__ ctrl) {
  const int tid = threadIdx.x;
  if (tid == 0) { ((unsigned*)ctrl)[0] = 0u; ctrl[1] = K; }
  __syncthreads();
  for (int pass = 3; pass >= 0; --pass) {
    const int shift = pass * 8;
    if (tid < 256) hist[tid] = 0u;
    __syncthreads();
    const unsigned pfx = ((unsigned*)ctrl)[0];
    for (int q = tid; q < N; q += BLOCK) {
      unsigned k = keys[q];
      bool match = (pass == 3) || ((k >> (shift + 8)) == (pfx >> (shift + 8)));
      if (match) atomicAdd(&hist[(k >> shift) & 255u], 1u);
    }
    __syncthreads();
    if (tid == 0) {
      int kk = ctrl[1];
      int cum = 0;
      unsigned bsel = 0u;
      for (int bbin = 255; bbin >= 0; --bbin) {
        int c = (int)hist[bbin];
        if (cum + c >= kk) { bsel = (unsigned)bbin; ctrl[1] = kk - cum; break; }
        cum += c;
      }
      ((unsigned*)ctrl)[0] = pfx | (bsel << shift);
    }
    __syncthreads();
  }
}

// ---------------------------------------------------------------------------
__global__ __launch_bounds__(BLOCK)
void path_quality_kernel(
    const float* __restrict__ x,
    const float* __restrict__ W0, const float* __restrict__ s0w1,
    const float* __restrict__ s0b1, const float* __restrict__ s0w2,
    const float* __restrict__ s0b2,
    const float* __restrict__ W1, const float* __restrict__ s1w1,
    const float* __restrict__ s1b1, const float* __restrict__ s1w2,
    const float* __restrict__ s1b2,
    const float* __restrict__ W2, const float* __restrict__ s2w1,
    const float* __restrict__ s2b1, const float* __restrict__ s2w2,
    const float* __restrict__ s2b2,
    float* __restrict__ out) {
  extern __shared__ unsigned char smem[];
  float*    Ct   = (float*)(smem + OFF_CT);
  unsigned* keys = (unsigned*)(smem + OFF_KEYS);
  float*    b1s  = (float*)(smem + OFF_B1);
  float*    w2s  = (float*)(smem + OFF_W2S);
  float*    pval = (float*)(smem + OFF_PVAL);
  float*    nval = (float*)(smem + OFF_NVAL);
  float*    nsc  = (float*)(smem + OFF_NSC);
  float*    g    = (float*)(smem + OFF_G);
  int*      pidx = (int*)(smem + OFF_PIDX);
  int*      nidx = (int*)(smem + OFF_NIDX);
  unsigned* hist = (unsigned*)(smem + OFF_HIST);
  int*      ctrl = (int*)(smem + OFF_CTRL);

  const int tid = threadIdx.x;
  const int b   = blockIdx.x;
  const float* xb = x + b * NFEAT;

  // ========== Layer 0: candidates i in [0,256), scalar d = x[b,i] ==========
  compute_Ct(s0w1, W0, NFEAT, Ct);
  if (tid < 32) { b1s[tid] = s0b1[tid]; w2s[tid] = s0w2[tid]; }
  __syncthreads();
  {
    const float b2 = s0b2[0];
    if (tid < NFEAT)
      keys[tid] = key_of(mlp_score(&Ct[tid * 32], xb[tid], b1s, w2s, b2));
  }
  __syncthreads();
  radix_select(keys, NFEAT, NPATH, hist, ctrl);
  if (tid == 0) { ctrl[2] = 0; ctrl[3] = 0; }
  __syncthreads();
  {
    const unsigned T = ((unsigned*)ctrl)[0];
    const int mt = ctrl[1];
    for (int q = tid; q < NFEAT; q += BLOCK) {
      unsigned k = keys[q];
      bool take = (k > T);
      if (!take && k == T) take = (atomicAdd(&ctrl[2], 1) < mt);
      if (take) {
        int slot = atomicAdd(&ctrl[3], 1);
        pidx[slot] = q;
        pval[slot] = xb[q];   // tanh folded into next-layer candidate formula
      }
    }
  }
  __syncthreads();

  // ========== Layer 1: 25600 candidates, d = tanh(W0[i, ip] * vp) ==========
  compute_Ct(s1w1, W1, NFEAT, Ct);
  if (tid < 32) { b1s[tid] = s1b1[tid]; w2s[tid] = s1w2[tid]; }
  __syncthreads();
  {
    const float b2 = s1b2[0];
    // q = i*100 + p: lanes in a wave share feature i -> Ct reads broadcast
    for (int q = tid; q < NCAND; q += BLOCK) {
      int i = q / NPATH;
      int p = q - i * NPATH;
      float d = tanhf(W0[i * NFEAT + pidx[p]] * pval[p]);
      keys[q] = key_of(mlp_score(&Ct[i * 32], d, b1s, w2s, b2));
    }
  }
  __syncthreads();
  radix_select(keys, NCAND, NPATH, hist, ctrl);
  if (tid == 0) { ctrl[2] = 0; ctrl[3] = 0; }
  __syncthreads();
  {
    const unsigned T = ((unsigned*)ctrl)[0];
    const int mt = ctrl[1];
    for (int q = tid; q < NCAND; q += BLOCK) {
      unsigned k = keys[q];
      bool take = (k > T);
      if (!take && k == T) take = (atomicAdd(&ctrl[2], 1) < mt);
      if (take) {
        int slot = atomicAdd(&ctrl[3], 1);
        int i = q / NPATH;
        int p = q - i * NPATH;
        nidx[slot] = i;
        nval[slot] = tanhf(W0[i * NFEAT + pidx[p]] * pval[p]);
      }
    }
  }
  __syncthreads();
  if (tid < NPATH) { pidx[tid] = nidx[tid]; pval[tid] = nval[tid]; }
  __syncthreads();

  // ========== Layer 2: 25600 candidates, d = tanh(W1[i, ip] * vp) ==========
  compute_Ct(s2w1, W2, FOUT_LAST, Ct);
  if (tid < 32) { b1s[tid] = s2b1[tid]; w2s[tid] = s2w2[tid]; }
  __syncthreads();
  {
    const float b2 = s2b2[0];
    for (int q = tid; q < NCAND; q += BLOCK) {
      int i = q / NPATH;
      int p = q - i * NPATH;
      float d = tanhf(W1[i * NFEAT + pidx[p]] * pval[p]);
      keys[q] = key_of(mlp_score(&Ct[i * 32], d, b1s, w2s, b2));
    }
  }
  __syncthreads();
  radix_select(keys, NCAND, NPATH, hist, ctrl);
  if (tid == 0) { ctrl[2] = 0; ctrl[3] = 0; }
  __syncthreads();
  {
    const unsigned T = ((unsigned*)ctrl)[0];
    const int mt = ctrl[1];
    for (int q = tid; q < NCAND; q += BLOCK) {
      unsigned k = keys[q];
      bool take = (k > T);
      if (!take && k == T) take = (atomicAdd(&ctrl[2], 1) < mt);
      if (take) {
        int slot = atomicAdd(&ctrl[3], 1);
        int i = q / NPATH;
        int p = q - i * NPATH;
        nidx[slot] = i;                               // final feature index i2
        nval[slot] = tanhf(W1[i * NFEAT + pidx[p]] * pval[p]);  // v2 (no tanh after W2)
        nsc[slot]  = val_of(k);   // final score == selection score (prune = column pick)
      }
    }
  }
  __syncthreads();

  // ========== Output: out[b,:] = W2 @ g,  g[i] = sum softmax_w * v2 =========
  if (tid < NFEAT) g[tid] = 0.0f;
  __syncthreads();
  float m = -INFINITY;
  for (int q = 0; q < NPATH; ++q) m = fmaxf(m, nsc[q]);
  float Z = 0.0f;
  for (int q = 0; q < NPATH; ++q) Z += expf(nsc[q] - m);
  if (tid < NPATH) {
    float w = expf(nsc[tid] - m) / Z;
    atomicAdd(&g[nidx[tid]], w * nval[tid]);
  }
  __syncthreads();
  if (tid < FOUT_LAST) {
    const float* wrow = W2 + tid * NFEAT;
    float acc = 0.0f;
#pragma unroll 8
    for (int i = 0; i < NFEAT; ++i) acc = fmaf(wrow[i], g[i], acc);
    out[b * FOUT_LAST + tid] = acc;
  }
}

// ---------------------------------------------------------------------------
extern "C" void kernel_launch(void* const* d_in, const int* in_sizes, int n_in,
                              void* d_out, int out_size, void* d_ws, size_t ws_size,
                              hipStream_t stream) {
  (void)in_sizes; (void)n_in; (void)out_size; (void)d_ws; (void)ws_size;
  const float* x    = (const float*)d_in[0];
  const float* W0   = (const float*)d_in[1];
  const float* s0w1 = (const float*)d_in[2];
  const float* s0b1 = (const float*)d_in[3];
  const float* s0w2 = (const float*)d_in[4];
  const float* s0b2 = (const float*)d_in[5];
  const float* W1   = (const float*)d_in[6];
  const float* s1w1 = (const float*)d_in[7];
  const float* s1b1 = (const float*)d_in[8];
  const float* s1w2 = (const float*)d_in[9];
  const float* s1b2 = (const float*)d_in[10];
  const float* W2   = (const float*)d_in[11];
  const float* s2w1 = (const float*)d_in[12];
  const float* s2b1 = (const float*)d_in[13];
  const float* s2w2 = (const float*)d_in[14];
  const float* s2b2 = (const float*)d_in[15];
  float* out = (float*)d_out;

  (void)hipFuncSetAttribute((const void*)path_quality_kernel,
                            hipFuncAttributeMaxDynamicSharedMemorySize,
                            SMEM_BYTES);
  path_quality_kernel<<<dim3(16), dim3(BLOCK), SMEM_BYTES, stream>>>(
      x, W0, s0w1, s0b1, s0w2, s0b2,
      W1, s1w1, s1b1, s1w2, s1b2,
      W2, s2w1, s2b1, s2w2, s2b2, out);
}